// SRP_map_9242769622023
// MI455X (gfx1250) — compile-verified
//
#include <hip/hip_runtime.h>
#include <hip/hip_bf16.h>
#include <math.h>

// ---------------- problem constants (from reference) ----------------
#define NPAIR 144          // N*N mic pairs
#define KLEN  4096         // GCC window
#define GPTS  2048         // RES_T * RES_P
#define PC    4            // pair-rows staged per chunk
#define NCHUNK (NPAIR / PC)        // 36
#define BLOCK 256
#define GSLICE 512                 // grid points per block (4 slices cover 2048)
#define SEG_PER_TH ((PC * KLEN * 4 / 16) / BLOCK)   // 16 b128 async ops / thread / chunk

typedef float v2f __attribute__((ext_vector_type(2)));
typedef float v8f __attribute__((ext_vector_type(8)));

#if !__has_builtin(__builtin_amdgcn_wmma_f32_16x16x4_f32)
#define NO_WMMA_F32X4 1
#endif

// ---- CDNA5 async copy: global -> LDS, 16B per lane, tracked by ASYNCcnt ----
__device__ __forceinline__ void async_b128(unsigned lds_byte, unsigned goff,
                                           const float* sbase) {
    asm volatile("global_load_async_to_lds_b128 %0, %1, %2"
                 :: "v"(lds_byte), "v"(goff), "s"(sbase)
                 : "memory");
}

template <int N>
__device__ __forceinline__ void wait_async() {
    asm volatile("s_wait_asynccnt %0" :: "n"(N) : "memory");
}

// =====================================================================
// Kernel 1: gather + pair-sum.  One block = (frame, 512-point g-slice).
// Pair reduction is done on the matrix pipe: ones[16x4] @ gathered[4x16]
// via V_WMMA_F32_16X16X4_F32, accumulating C across 36 chunks.
// =====================================================================
__global__ __launch_bounds__(BLOCK) void srp_accum(const float* __restrict__ x,
                                                   const int* __restrict__ tau0,
                                                   float* __restrict__ out) {
    __shared__ float lds_x[2 * PC * KLEN];   // 128 KB, double-buffered

    const int tid  = threadIdx.x;
    const int lane = tid & 31;
    const int wave = tid >> 5;               // 0..7
    const int f    = blockIdx.y;             // frame
    const int gsl  = blockIdx.x * GSLICE;    // g-slice base

    const float* xf = x + (size_t)f * (NPAIR * KLEN);
    const unsigned lds0 = (unsigned)(uintptr_t)lds_x;  // LDS byte offset (low 32b of generic ptr)

    const int hi = lane >> 4;                // B-matrix half: lanes 16-31 carry K=2,3
    const int gl = lane & 15;                // column within 16-wide g-tile

    int gcol[4];
#pragma unroll
    for (int i = 0; i < 4; ++i) gcol[i] = gsl + wave * 64 + i * 16 + gl;

    v8f acc[4];
#pragma unroll
    for (int i = 0; i < 4; ++i)
#pragma unroll
        for (int e = 0; e < 8; ++e) acc[i][e] = 0.0f;

    // prefetch chunk 0
    {
        const float* gsrc = xf;
        const unsigned base = lds0;
#pragma unroll
        for (int i = 0; i < SEG_PER_TH; ++i) {
            unsigned off = (unsigned)(tid + i * BLOCK) * 16u;
            async_b128(base + off, off, gsrc);
        }
    }

    for (int c = 0; c < NCHUNK; ++c) {
        if (c + 1 < NCHUNK) {
            // issue next chunk into the other buffer (its last readers finished
            // before the barrier that ended iteration c-1)
            const float* gsrc = xf + (size_t)(c + 1) * (PC * KLEN);
            const unsigned base = lds0 + (unsigned)(((c + 1) & 1) * PC * KLEN * 4);
#pragma unroll
            for (int i = 0; i < SEG_PER_TH; ++i) {
                unsigned off = (unsigned)(tid + i * BLOCK) * 16u;
                async_b128(base + off, off, gsrc);
            }
            wait_async<SEG_PER_TH>();   // older 16 (chunk c) retired; chunk c+1 in flight
        } else {
            wait_async<0>();
        }
        __syncthreads();                 // chunk c visible to all waves

        // this lane's two K-rows for the WMMA B operand
        const int    prow = c * PC + hi * 2;             // global pair row (K0 of this half)
        const int*   t0   = tau0 + (size_t)prow * GPTS;  // lag table row j
        const int*   t1   = t0 + GPTS;                   // row j+1
        const float* buf  = lds_x + (c & 1) * (PC * KLEN);
        const float* b0   = buf + (hi * 2) * KLEN;       // staged row j   (LDS)
        const float* b1   = b0 + KLEN;                   // staged row j+1 (LDS)

#pragma unroll
        for (int i = 0; i < 4; ++i) {
            const int g  = gcol[i];
            const int i0 = t0[g];
            const int i1 = t1[g];
            v2f bmat;                    // VGPR0: K0|K2, VGPR1: K1|K3 (32-bit B striping)
            bmat.x = b0[i0];             // ds_load_b32 gather
            bmat.y = b1[i1];
#ifndef NO_WMMA_F32X4
            v2f amat; amat.x = 1.0f; amat.y = 1.0f;   // ones => D = column sums
            acc[i] = __builtin_amdgcn_wmma_f32_16x16x4_f32(
                false, amat, false, bmat, (short)0, acc[i], false, false);
#else
            acc[i][0] += bmat.x + bmat.y;             // scalar fallback (half-sums/lane)
#endif
        }
        __syncthreads();                 // buffer (c&1) free for chunk c+2
    }

    // D row 0: lane n holds sum over all 144 pairs for g-tile column n
#pragma unroll
    for (int i = 0; i < 4; ++i) {
        float v = acc[i][0];
#ifdef NO_WMMA_F32X4
        v += __shfl_xor(v, 16, 32);      // combine K01 half with K23 half
#endif
        if (lane < 16) out[(size_t)f * GPTS + gcol[i]] = v;
    }
}

// =====================================================================
// Kernel 2: per-frame normalization, in place in d_out.
// maps = maps - mean(maps); maps += 1e-12; maps /= max(maps)
// =====================================================================
__global__ __launch_bounds__(BLOCK) void srp_norm(float* __restrict__ m) {
    __shared__ float red[BLOCK];
    const int f = blockIdx.x;
    const int tid = threadIdx.x;
    float* p = m + (size_t)f * GPTS;

    float v[8];
    float s = 0.0f;
#pragma unroll
    for (int i = 0; i < 8; ++i) { v[i] = p[tid + i * BLOCK]; s += v[i]; }

    red[tid] = s;
    __syncthreads();
    for (int st = BLOCK / 2; st > 0; st >>= 1) {
        if (tid < st) red[tid] += red[tid + st];
        __syncthreads();
    }
    const float mean = red[0] * (1.0f / GPTS);
    __syncthreads();

    float mx = -INFINITY;
#pragma unroll
    for (int i = 0; i < 8; ++i) {
        v[i] = v[i] - mean + 1e-12f;
        mx = fmaxf(mx, v[i]);
    }
    red[tid] = mx;
    __syncthreads();
    for (int st = BLOCK / 2; st > 0; st >>= 1) {
        if (tid < st) red[tid] = fmaxf(red[tid], red[tid + st]);
        __syncthreads();
    }
    const float inv = 1.0f / red[0];
#pragma unroll
    for (int i = 0; i < 8; ++i) p[tid + i * BLOCK] = v[i] * inv;
}

// =====================================================================
extern "C" void kernel_launch(void* const* d_in, const int* in_sizes, int n_in,
                              void* d_out, int out_size, void* d_ws, size_t ws_size,
                              hipStream_t stream) {
    (void)n_in; (void)out_size; (void)d_ws; (void)ws_size;
    const float* x    = (const float*)d_in[0];   // [B,T,N,N,K] f32
    const int*   tau0 = (const int*)d_in[1];     // [N,N,Th,Ph] i32
    float*       out  = (float*)d_out;           // [B,T,Th,Ph] f32

    const int frames = in_sizes[0] / (NPAIR * KLEN);   // B*T = 200

    dim3 grid(GPTS / GSLICE, frames);                  // (4, 200)
    srp_accum<<<grid, BLOCK, 0, stream>>>(x, tau0, out);
    srp_norm<<<frames, BLOCK, 0, stream>>>(out);
}